// MI_69999376990883
// MI455X (gfx1250) — compile-verified
//
#include <hip/hip_runtime.h>
#include <hip/hip_bf16.h>
#include <stdint.h>

#define NBINS 256
#define BLOCK 1024
#define CHUNK 49152                          // elements/chunk; <=65535 -> u16-packed counts cannot overflow
#define VITERS (CHUNK / (BLOCK * 4))         // 12 float4 iterations per chunk
#define PACKED_J (NBINS * NBINS / 2)         // 32768 u32 words per packed 256x256 joint
#define HIST_WORDS (3 * NBINS)               // 768
#define WPB (2 * PACKED_J + HIST_WORDS)      // 66304 u32 words per partial slot (259 KB)
#define FIN_JOINT (2 * NBINS * NBINS)        // 131072
#define FIN_WORDS (FIN_JOINT + HIST_WORDS)   // 131840

typedef float v4f __attribute__((ext_vector_type(4)));   // native vector: OK for nontemporal builtin

__device__ __forceinline__ int binof(float x, bool& in) {
    // np.histogram(range=(0,1)): include [0,1], rightmost edge closed
    in = (x >= 0.0f) && (x <= 1.0f);
    int i = (int)floorf(x * 256.0f);
    i = i < 0 ? 0 : i;
    i = i > 255 ? 255 : i;
    return i;
}

__global__ __launch_bounds__(64) void mi_zero_sums(unsigned* sums) {
    if (threadIdx.x < 8) sums[threadIdx.x] = 0u;
}

__device__ __forceinline__ void mi_accum(float fx, float ax, float bx,
                                         unsigned* jAF, unsigned* jBF,
                                         unsigned* hF, unsigned* hA, unsigned* hB) {
    bool inF, inA, inB;
    int iF = binof(fx, inF);
    int iA = binof(ax, inA);
    int iB = binof(bx, inB);
    if (inF) atomicAdd(&hF[iF], 1u);
    if (inA) atomicAdd(&hA[iA], 1u);
    if (inB) atomicAdd(&hB[iB], 1u);
    if (inF && inA) {
        int flat = (iF << 8) | iA;              // row = F bin, col = A bin
        atomicAdd(&jAF[flat >> 1], 1u << ((flat & 1) << 4));
    }
    if (inF && inB) {
        int flat = (iF << 8) | iB;
        atomicAdd(&jBF[flat >> 1], 1u << ((flat & 1) << 4));
    }
}

// One pass over F/A/B. Each block privatizes BOTH joints (u16-packed) + all three
// 1D hists in 259 KB LDS (CDNA5: 320 KB/workgroup). Inputs are streamed with
// non-temporal loads so the 151 MB one-shot stream does not evict the L2-resident
// partials. Flush goes out via the gfx1250 async LDS->global DMA path.
__global__ __launch_bounds__(BLOCK) void mi_hist(const float* __restrict__ F,
                                                 const float* __restrict__ A,
                                                 const float* __restrict__ B,
                                                 unsigned* __restrict__ partials,
                                                 int n, int nchunks) {
    __shared__ __align__(16) unsigned lds[WPB];
    unsigned* jAF = lds;
    unsigned* jBF = lds + PACKED_J;
    unsigned* hF  = lds + 2 * PACKED_J;
    unsigned* hA  = hF + NBINS;
    unsigned* hB  = hA + NBINS;

    for (int i = threadIdx.x; i < WPB; i += BLOCK) lds[i] = 0u;
    __syncthreads();

    const int n4 = n >> 2;
    const v4f* F4 = (const v4f*)F;
    const v4f* A4 = (const v4f*)A;
    const v4f* B4 = (const v4f*)B;

    // Normally one chunk per block (exact u16 safety). If scratch forced fewer
    // slots, a block accumulates several chunks into the same private LDS hist.
    for (int c = blockIdx.x; c < nchunks; c += gridDim.x) {
        const int vbase = (c * CHUNK) >> 2;
        #pragma unroll 1
        for (int it = 0; it < VITERS; it += 2) {      // 2-deep SW pipeline: 6 B128 loads in flight
            int v0 = vbase + it * BLOCK + threadIdx.x;
            int v1 = v0 + BLOCK;
            bool ok0 = v0 < n4, ok1 = v1 < n4;
            v4f f0 = {}, a0 = {}, b0 = {}, f1 = {}, a1 = {}, b1 = {};
            if (ok0) {
                f0 = __builtin_nontemporal_load(F4 + v0);   // global_load_b128 th:NT
                a0 = __builtin_nontemporal_load(A4 + v0);
                b0 = __builtin_nontemporal_load(B4 + v0);
            }
            if (ok1) {
                f1 = __builtin_nontemporal_load(F4 + v1);
                a1 = __builtin_nontemporal_load(A4 + v1);
                b1 = __builtin_nontemporal_load(B4 + v1);
            }
            if (ok0) {
                mi_accum(f0.x, a0.x, b0.x, jAF, jBF, hF, hA, hB);
                mi_accum(f0.y, a0.y, b0.y, jAF, jBF, hF, hA, hB);
                mi_accum(f0.z, a0.z, b0.z, jAF, jBF, hF, hA, hB);
                mi_accum(f0.w, a0.w, b0.w, jAF, jBF, hF, hA, hB);
            }
            if (ok1) {
                mi_accum(f1.x, a1.x, b1.x, jAF, jBF, hF, hA, hB);
                mi_accum(f1.y, a1.y, b1.y, jAF, jBF, hF, hA, hB);
                mi_accum(f1.z, a1.z, b1.z, jAF, jBF, hF, hA, hB);
                mi_accum(f1.w, a1.w, b1.w, jAF, jBF, hF, hA, hB);
            }
        }
    }
    __syncthreads();

    // Bulk flush: 259 KB LDS -> this block's slot. CDNA5 async DMA (ASYNCcnt-
    // tracked, no VGPR round trip); stores default-RT so partials sit in L2
    // (<= 66 MB << 192 MB) for the reduce pass.
    unsigned lbase = (unsigned)(uintptr_t)&lds[0];
    unsigned long long gbase =
        (unsigned long long)(uintptr_t)(partials + (size_t)blockIdx.x * WPB);
    for (unsigned i = threadIdx.x; i < WPB / 4; i += BLOCK) {
        unsigned laddr = lbase + i * 16u;
        unsigned long long gaddr = gbase + (unsigned long long)i * 16u;
        asm volatile("global_store_async_from_lds_b128 %0, %1, off"
                     :: "v"(gaddr), "v"(laddr) : "memory");
    }
    asm volatile("s_wait_asynccnt 0x0" ::: "memory");
}

// Sum per-slot partials into final u32 counts; accumulate the 5 totals.
// Region boundaries (32768/32768/256/256/256 words) align with 256-thread blocks,
// so each block feeds exactly one of sums[0..4].
__global__ __launch_bounds__(256) void mi_reduce(const unsigned* __restrict__ partials,
                                                 unsigned* __restrict__ fin,
                                                 unsigned* __restrict__ sums,
                                                 int nslots) {
    __shared__ unsigned acc[256];
    int g = blockIdx.x * 256 + threadIdx.x;
    unsigned local = 0;
    if (g < 2 * PACKED_J) {                      // packed joint words (AF then BF)
        unsigned lo = 0, hi = 0;
        for (int b = 0; b < nslots; ++b) {
            unsigned w = partials[(size_t)b * WPB + g];
            lo += w & 0xFFFFu;
            hi += w >> 16;
        }
        fin[2 * g]     = lo;                     // AF -> fin[0,65536), BF -> [65536,131072)
        fin[2 * g + 1] = hi;
        local = lo + hi;
    } else if (g < WPB) {                        // plain u32 1D hist words
        unsigned s = 0;
        for (int b = 0; b < nslots; ++b) s += partials[(size_t)b * WPB + g];
        fin[FIN_JOINT + (g - 2 * PACKED_J)] = s;
        local = s;
    }
    acc[threadIdx.x] = local;
    __syncthreads();
    for (int off = 128; off > 0; off >>= 1) {
        if (threadIdx.x < off) acc[threadIdx.x] += acc[threadIdx.x + off];
        __syncthreads();
    }
    if (threadIdx.x == 0) {
        int cat = (blockIdx.x < 128) ? 0
                : (blockIdx.x < 256) ? 1
                : (2 + ((int)blockIdx.x - 256));  // 2=F, 3=A, 4=B
        atomicAdd(&sums[cat], acc[0]);
    }
}

// Final MI log-sum; faithful to the reference's marginal indexing quirk:
// term[i,j] = p_XF[i,j] * log((p_XF[i,j]+eps) / ((p_X[i]+eps)*(p_F[j]+eps)))
// with i = F-bin (row), j = A/B-bin (col). Counts < 2^24 so f32 is exact.
__global__ __launch_bounds__(BLOCK) void mi_final(const unsigned* __restrict__ fin,
                                                  const unsigned* __restrict__ sums,
                                                  float* __restrict__ out) {
    __shared__ float red[BLOCK];
    const float eps = 1e-10f;
    const float invAF = 1.0f / (float)sums[0];
    const float invBF = 1.0f / (float)sums[1];
    const float invF  = 1.0f / (float)sums[2];
    const float invA  = 1.0f / (float)sums[3];
    const float invB  = 1.0f / (float)sums[4];
    const unsigned* jAF = fin;
    const unsigned* jBF = fin + NBINS * NBINS;
    const unsigned* hF  = fin + FIN_JOINT;
    const unsigned* hA  = hF + NBINS;
    const unsigned* hB  = hA + NBINS;
    float acc = 0.0f;
    for (int k = threadIdx.x; k < NBINS * NBINS; k += BLOCK) {
        int i = k >> 8;     // F-bin (row)
        int j = k & 255;    // A/B-bin (col)
        float pAF = (float)jAF[k] * invAF;
        float pBF = (float)jBF[k] * invBF;
        float pFj = (float)hF[j] * invF;
        float pAi = (float)hA[i] * invA;
        float pBi = (float)hB[i] * invB;
        acc += pAF * logf((pAF + eps) / ((pAi + eps) * (pFj + eps)));
        acc += pBF * logf((pBF + eps) / ((pBi + eps) * (pFj + eps)));
    }
    red[threadIdx.x] = acc;
    __syncthreads();
    for (int off = BLOCK / 2; off > 0; off >>= 1) {
        if (threadIdx.x < off) red[threadIdx.x] += red[threadIdx.x + off];
        __syncthreads();
    }
    if (threadIdx.x == 0) out[0] = -red[0];
}

extern "C" void kernel_launch(void* const* d_in, const int* in_sizes, int n_in,
                              void* d_out, int out_size, void* d_ws, size_t ws_size,
                              hipStream_t stream) {
    const float* F = (const float*)d_in[0];
    const float* A = (const float*)d_in[1];
    const float* B = (const float*)d_in[2];
    const int n = in_sizes[0];                        // 16*3*512*512 = 12,582,912
    const int nchunks = (n + CHUNK - 1) / CHUNK;      // 256

    // Adapt partial-slot count to available scratch (fast path: one slot/chunk).
    size_t ws_words = ws_size / 4;
    long avail = (long)ws_words - (long)FIN_WORDS - 64;
    int nslots = (avail > 0) ? (int)(avail / WPB) : 0;
    if (nslots < 1) nslots = 1;
    if (nslots > nchunks) nslots = nchunks;

    unsigned* P    = (unsigned*)d_ws;                          // per-slot partials
    unsigned* FINp = P + (size_t)nslots * WPB;                 // final counts
    unsigned* SUMS = FINp + FIN_WORDS;                         // 5 totals

    mi_zero_sums<<<1, 64, 0, stream>>>(SUMS);
    mi_hist<<<nslots, BLOCK, 0, stream>>>(F, A, B, P, n, nchunks);
    const int rblocks = (WPB + 255) / 256;                     // 259
    mi_reduce<<<rblocks, 256, 0, stream>>>(P, FINp, SUMS, nslots);
    mi_final<<<1, BLOCK, 0, stream>>>(FINp, SUMS, (float*)d_out);
}